// UpSample_83227876262280
// MI455X (gfx1250) — compile-verified
//
#include <hip/hip_runtime.h>
#include <hip/hip_bf16.h>

// Problem sizes from the reference: B=8, n=8192, m=2048, C=128
#define BB 8
#define NN 8192
#define MM 2048
#define CC 128

typedef __attribute__((ext_vector_type(2))) float v2f;
typedef __attribute__((ext_vector_type(8))) float v8f;

// ---------------------------------------------------------------------------
// Kernel 1: 1-NN index via WMMA f32 16x16x4.
//   d'(i,j) = |k_j|^2 - 2 u_i . k_j   (argmin-equivalent to full sq. distance)
//   A row  = [-2ux, -2uy, -2uz, 1]   (16x4)
//   B col  = [ kx,   ky,   kz, |k|^2] (4x16)
// Block = 256 threads = 8 waves; each wave owns one 16-row query tile.
// The 2048-point known table (float4: x,y,z,|k|^2 = 32KB) is staged in LDS
// once per block; the sweep is unrolled 4x so ds_load latency overlaps the
// WMMA + argmin-update work of the previous tiles.
// ---------------------------------------------------------------------------
__global__ __launch_bounds__(256) void nn_idx_kernel(
    const float* __restrict__ unknown,   // (B, N, 3)
    const float* __restrict__ known,     // (B, M, 3)
    int* __restrict__ idx_out)           // (B, N)
{
    __shared__ float4 kn[MM];            // 32 KB of 320 KB/WGP

    const int b        = blockIdx.x >> 6;          // 64 blocks per batch
    const int rowBlock = (blockIdx.x & 63) * 128;  // 8 waves * 16 rows
    const int tid      = threadIdx.x;

    // Stage known points + squared norms into LDS (8 points per thread).
    const float* kb = known + b * (MM * 3);
    for (int j = tid; j < MM; j += 256) {
        float x = kb[j * 3 + 0];
        float y = kb[j * 3 + 1];
        float z = kb[j * 3 + 2];
        kn[j] = make_float4(x, y, z, x * x + y * y + z * z);
    }
    __syncthreads();

    const int wave = tid >> 5;
    const int lane = tid & 31;
    const int col  = lane & 15;   // N (column) / M (A-row) sub-index
    const int half = lane >> 4;   // K-half selector

    // A-matrix fragment: row M = col, lanes 0-15 carry K={0,1}, 16-31 K={2,3}
    const int row = rowBlock + wave * 16 + col;
    const float* u = unknown + (b * NN + row) * 3;
    const float ux = u[0], uy = u[1], uz = u[2];
    v2f a;
    a.x = half ? (-2.0f * uz) : (-2.0f * ux);
    a.y = half ? 1.0f         : (-2.0f * uy);

    // Per-lane B fragment base: element j lives at float2 index j*2 + half.
    const float2* __restrict__ lp = (const float2*)kn + (col * 2 + half);

    float minv[8];
    int   mini[8];
#pragma unroll
    for (int r = 0; r < 8; ++r) { minv[r] = 3.402823466e38f; mini[r] = 0; }

    for (int jb = 0; jb < MM; jb += 64) {
        // Batch 4 ds_load_b64 (immediate offsets off one base) before use.
        float2 kv[4];
#pragma unroll
        for (int q = 0; q < 4; ++q) kv[q] = lp[(jb + 16 * q) * 2];

#pragma unroll
        for (int q = 0; q < 4; ++q) {
            v2f bv;
            bv.x = kv[q].x;
            bv.y = kv[q].y;

            v8f acc = {};
            acc = __builtin_amdgcn_wmma_f32_16x16x4_f32(
                /*neg_a=*/false, a, /*neg_b=*/false, bv,
                /*c_mod=*/(short)0, acc, /*reuse_a=*/false, /*reuse_b=*/false);

            const int j = jb + 16 * q + col;
#pragma unroll
            for (int r = 0; r < 8; ++r) {
                const float d = acc[r];
                // strict < keeps first occurrence (smallest j) like jnp.argmin
                const bool t = d < minv[r];
                minv[r] = t ? d : minv[r];
                mini[r] = t ? j : mini[r];
            }
        }
    }

    // Butterfly argmin across the 16 lanes of each half (wave32: offsets
    // 1,2,4,8 never cross the 16-lane boundary). Branch-free tie-break to
    // the smaller index (non-short-circuit ops -> pure v_cndmask).
#pragma unroll
    for (int off = 1; off < 16; off <<= 1) {
#pragma unroll
        for (int r = 0; r < 8; ++r) {
            const float ov = __shfl_xor(minv[r], off, 32);
            const int   oi = __shfl_xor(mini[r], off, 32);
            const bool  t  = (ov < minv[r]) | ((ov == minv[r]) & (oi < mini[r]));
            minv[r] = t ? ov : minv[r];
            mini[r] = t ? oi : mini[r];
        }
    }

    // Lane col==0 of each half holds rows M = r + 8*half of this tile.
    if (col == 0) {
        const int base = b * NN + rowBlock + wave * 16 + half * 8;
#pragma unroll
        for (int r = 0; r < 8; ++r) idx_out[base + r] = mini[r];
    }
}

// ---------------------------------------------------------------------------
// Kernel 2: feature gather. out[b][c][i] = kf[b][c][idx[b][i]].
// One thread per (b,i); idx read once, 128 coalesced channel writes; the
// gathered kf row is 8 KB -> stays resident in cache.
// ---------------------------------------------------------------------------
__global__ __launch_bounds__(256) void gather_feats_kernel(
    const float* __restrict__ kf,   // (B, C, M)
    const int*   __restrict__ idx,  // (B, N)
    float* __restrict__ out)        // (B, C, N)
{
    const int t = blockIdx.x * 256 + threadIdx.x;   // over B*N
    const int b = t / NN;
    const int i = t - b * NN;
    const int id = idx[t];
    const float* kfb = kf  + b * (CC * MM);
    float*       ob  = out + b * (CC * NN);
#pragma unroll 4
    for (int c = 0; c < CC; ++c) {
        ob[c * NN + i] = kfb[c * MM + id];
    }
}

extern "C" void kernel_launch(void* const* d_in, const int* in_sizes, int n_in,
                              void* d_out, int out_size, void* d_ws, size_t ws_size,
                              hipStream_t stream) {
    const float* unknown     = (const float*)d_in[0];  // (B, N, 3)
    const float* known       = (const float*)d_in[1];  // (B, M, 3)
    const float* known_feats = (const float*)d_in[2];  // (B, C, M)
    float* out = (float*)d_out;                        // (B, C, N)
    int* idx   = (int*)d_ws;                           // (B, N) = 256 KB scratch

    // Kernel 1: 64 blocks per batch * 8 batches = 512 blocks of 256 threads.
    nn_idx_kernel<<<BB * (NN / 128), 256, 0, stream>>>(unknown, known, idx);

    // Kernel 2: one thread per (b, i) -> 8*8192/256 = 256 blocks.
    gather_feats_kernel<<<(BB * NN) / 256, 256, 0, stream>>>(known_feats, idx, out);
}